// Net_24395414241687
// MI455X (gfx1250) — compile-verified
//
#include <hip/hip_runtime.h>
#include <hip/hip_bf16.h>

typedef __attribute__((ext_vector_type(2))) float v2f;
typedef __attribute__((ext_vector_type(8))) float v8f;

__device__ __forceinline__ float eluf(float x) {
    return x > 0.f ? x : (expf(x) - 1.f);
}

// ---------------------------------------------------------------------------
// Zero-init the atomic-accumulation region of the workspace.
// ---------------------------------------------------------------------------
__global__ void zero_kernel(float* __restrict__ p, int n) {
    int i = blockIdx.x * blockDim.x + threadIdx.x;
    if (i < n) p[i] = 0.f;
}

// ---------------------------------------------------------------------------
// One-time swizzle of the expanded edge-MLP weights (plus its output bias as
// a 26th slab) into WMMA B-fragment order, zero-padded to C_INP rows/slab:
//   Wt v2f element [((k*(C_INP/4) + i0/4)*2 + g)*C_OUT + col] =
//       { W[(k, i0+2g), col], W[(k, i0+2g+1), col] }
// so each lane's two B values are one aligned 8-byte load at a uniform stride.
// ---------------------------------------------------------------------------
template <int C_IN, int C_OUT>
__global__ void swizzle_wb_kernel(const float* __restrict__ Wb,   // [25, C_IN*C_OUT]
                                  const float* __restrict__ bb,   // [C_IN*C_OUT]
                                  float* __restrict__ Wt,         // [26*C_INP*C_OUT]
                                  int total) {
    constexpr int C_INP = (C_IN + 3) & ~3;
    constexpr int CKO   = C_IN * C_OUT;
    int idx = blockIdx.x * blockDim.x + threadIdx.x;
    if (idx >= total) return;
    int t    = idx;
    int half = t & 1;            t >>= 1;
    int col  = t % C_OUT;        t /= C_OUT;
    int g    = t & 1;            t >>= 1;
    int chunk= t % (C_INP / 4);  t /= (C_INP / 4);
    int k    = t;                // 0..25 (25 == bias slab)
    int row  = chunk * 4 + g * 2 + half;
    float v  = 0.f;
    if (row < C_IN) {
        v = (k < 25) ? Wb[(size_t)k * CKO + row * C_OUT + col]
                     : bb[row * C_OUT + col];
    }
    Wt[idx] = v;
}

// ---------------------------------------------------------------------------
// Per-edge 19->25 hidden layer of BOTH edge-MLPs (tiny: ~95M MAC total),
// plus the per-destination-node edge count (shared by both conv layers).
// ---------------------------------------------------------------------------
__global__ void edge_mlp_count_kernel(const float* __restrict__ EA,
                                      const float* __restrict__ W1a, const float* __restrict__ b1a,
                                      const float* __restrict__ W2a, const float* __restrict__ b2a,
                                      const int*   __restrict__ dst,
                                      float* __restrict__ T1, float* __restrict__ T2,
                                      float* __restrict__ cnt, int E) {
    int e = blockIdx.x * blockDim.x + threadIdx.x;
    if (e >= E) return;
    float ea[19];
#pragma unroll
    for (int i = 0; i < 19; ++i) ea[i] = EA[e * 19 + i];
    for (int j = 0; j < 25; ++j) {
        float t1 = b1a[j], t2 = b2a[j];
#pragma unroll
        for (int i = 0; i < 19; ++i) {
            t1 = fmaf(ea[i], W1a[i * 25 + j], t1);
            t2 = fmaf(ea[i], W2a[i * 25 + j], t2);
        }
        T1[e * 25 + j] = fmaxf(t1, 0.f);   // relu
        T2[e * 25 + j] = fmaxf(t2, 0.f);
    }
    atomicAdd(&cnt[dst[e]], 1.f);
}

// ---------------------------------------------------------------------------
// Fused NNConv message kernel (the hot loop), via V_WMMA_F32_16X16X4_F32.
//
//   msg[16 edges x C_OUT] = f[16 x K] @ W[K x C_OUT]
// with per-k-slab K ordering: f[e,(k,i)] = T[e,k]*h_src[e,i] (slab 25: T==1,
// folding the edge-MLP output bias). A fragments are built on the fly from
// LDS (zero-padded rows); B fragments come from the pre-swizzled Wt with one
// aligned global_load_b64 per WMMA at a compile-time stride.
//
// One wave per 16-wide output tile; C accumulator stays in 8 VGPRs and is
// scattered to agg[dst] with global f32 atomics (scatter-mean numerator).
// ---------------------------------------------------------------------------
template <int C_IN, int C_OUT>
__global__ void nnconv_msg_wmma_kernel(const float* __restrict__ h,    // [N, C_IN]
                                       const float* __restrict__ T,    // [E, 25]
                                       const float* __restrict__ Wt,   // swizzled [26*C_INP*C_OUT]
                                       const int*   __restrict__ src,
                                       const int*   __restrict__ dst,
                                       float* __restrict__ agg,        // [N, C_OUT] atomics
                                       int E) {
    constexpr int C_INP = (C_IN + 3) & ~3;   // i-range padded to multiple of 4
    constexpr int HS    = C_INP + 1;         // odd LDS stride -> conflict-free
    constexpr int WSTEP = 2 * C_OUT;         // v2f stride per 4-row K chunk

    __shared__ float h_s[16 * HS];
    __shared__ float T_s[16 * 25];
    __shared__ int   dst_s[16];

    const int ebase = blockIdx.x * 16;

    // warm L2 path for the (small, shared) swizzled weight matrix
    __builtin_prefetch(Wt + ((size_t)(threadIdx.x & 31) << 6), 0, 1);

    // Cooperative gather of the 16 source-node feature rows (zero-padded) + T.
    for (int idx = threadIdx.x; idx < 16 * C_INP; idx += blockDim.x) {
        int le = idx / C_INP, i = idx % C_INP;
        int ee = min(ebase + le, E - 1);
        h_s[le * HS + i] = (i < C_IN) ? h[(size_t)src[ee] * C_IN + i] : 0.f;
    }
    for (int idx = threadIdx.x; idx < 16 * 25; idx += blockDim.x) {
        int le = idx / 25, j = idx % 25;
        int ee = min(ebase + le, E - 1);
        T_s[le * 25 + j] = T[ee * 25 + j];
    }
    if (threadIdx.x < 16) dst_s[threadIdx.x] = dst[min(ebase + (int)threadIdx.x, E - 1)];
    __syncthreads();

    const int lane  = threadIdx.x & 31;
    const int otile = threadIdx.x >> 5;       // one 16-col N tile per wave
    const int m     = lane & 15;              // A row (edge) / B col within tile
    const int kb    = (lane >> 4) << 1;       // lane-group K offset {0,2}
    const int col   = otile * 16 + m;         // global output channel

    const float* hrow = h_s + m * HS;         // this lane's A-row in LDS
    const float* trow = T_s + m * 25;
    // this lane's B stream: one v2f per K-chunk, uniform stride WSTEP
    const v2f* wl = reinterpret_cast<const v2f*>(Wt) + ((lane >> 4) * C_OUT + col);

    v8f c = {0.f, 0.f, 0.f, 0.f, 0.f, 0.f, 0.f, 0.f};

    for (int k = 0; k < 25; ++k) {
        const float tk = trow[k];                          // 1 LDS read per slab
        const v2f* wk  = wl + (size_t)k * (C_INP / 4) * WSTEP;
#pragma unroll 4
        for (int i0 = 0; i0 < C_INP; i0 += 4) {
            const int ii = i0 + kb;                 // this lane-group's K rows
            v2f a;
            a.x = tk * hrow[ii];                    // consecutive -> ds pair load
            a.y = tk * hrow[ii + 1];
            const v2f b = wk[(i0 >> 2) * WSTEP];    // global_load_b64, imm offset
            c = __builtin_amdgcn_wmma_f32_16x16x4_f32(false, a, false, b,
                                                      (short)0, c, false, false);
        }
    }
    // bias slab (slab 25 of Wt): f rows are h_src itself (T == 1)
    {
        const v2f* wk = wl + (size_t)25 * (C_INP / 4) * WSTEP;
#pragma unroll 4
        for (int i0 = 0; i0 < C_INP; i0 += 4) {
            const int ii = i0 + kb;
            v2f a;
            a.x = hrow[ii];
            a.y = hrow[ii + 1];
            const v2f b = wk[(i0 >> 2) * WSTEP];
            c = __builtin_amdgcn_wmma_f32_16x16x4_f32(false, a, false, b,
                                                      (short)0, c, false, false);
        }
    }

    // Scatter-add the accumulator to agg[dst]: VGPR j, lane l -> row j+8*(l>>4)
    const int rowbase = (lane >> 4) * 8;
#pragma unroll
    for (int j = 0; j < 8; ++j) {
        const int le = rowbase + j;
        if (ebase + le < E) {
            atomicAdd(&agg[(size_t)dst_s[le] * C_OUT + col], c[j]);
        }
    }
}

// ---------------------------------------------------------------------------
// h_out = elu(h_in @ root + agg/cnt + bias)   (root matmuls are tiny)
// ---------------------------------------------------------------------------
template <int C_IN, int C_OUT>
__global__ void node_update_kernel(const float* __restrict__ hin,
                                   const float* __restrict__ root,
                                   const float* __restrict__ bias,
                                   const float* __restrict__ agg,
                                   const float* __restrict__ cnt,
                                   float* __restrict__ hout, int Nn) {
    int idx = blockIdx.x * blockDim.x + threadIdx.x;
    if (idx >= Nn * C_OUT) return;
    int n = idx / C_OUT, o = idx % C_OUT;
    float acc = bias[o];
    const float* hr = hin + (size_t)n * C_IN;
    for (int i = 0; i < C_IN; ++i) acc = fmaf(hr[i], root[i * C_OUT + o], acc);
    acc += agg[idx] / fmaxf(cnt[n], 1.f);
    hout[idx] = eluf(acc);
}

// ---------------------------------------------------------------------------
// Global mean-pool numerator/denominator.
// ---------------------------------------------------------------------------
__global__ void pool_kernel(const float* __restrict__ h2, const int* __restrict__ batch,
                            float* __restrict__ gsum, float* __restrict__ gcnt, int Nn) {
    int idx = blockIdx.x * blockDim.x + threadIdx.x;
    if (idx >= Nn * 64) return;
    int n = idx >> 6, o = idx & 63;
    int b = batch[n];
    atomicAdd(&gsum[b * 64 + o], h2[idx]);
    if (o == 0) atomicAdd(&gcnt[b], 1.f);
}

// ---------------------------------------------------------------------------
// Readout: elu(fc1) -> fc2 -> log_softmax over a size-1 axis (== 0, but
// computed via the stable formula so NaN propagation matches the reference).
// ---------------------------------------------------------------------------
__global__ void readout_kernel(const float* __restrict__ gsum, const float* __restrict__ gcnt,
                               const float* __restrict__ fc1W, const float* __restrict__ fc1b,
                               const float* __restrict__ fc2W, const float* __restrict__ fc2b,
                               float* __restrict__ out) {
    __shared__ float gm[64];
    __shared__ float red[128];
    const int g = blockIdx.x, t = threadIdx.x;
    if (t < 64) gm[t] = gsum[g * 64 + t] / fmaxf(gcnt[g], 1.f);
    __syncthreads();
    float acc = fc1b[t];
    for (int o = 0; o < 64; ++o) acc = fmaf(gm[o], fc1W[o * 128 + t], acc);
    red[t] = eluf(acc) * fc2W[t];
    __syncthreads();
    for (int s = 64; s > 0; s >>= 1) {
        if (t < s) red[t] += red[t + s];
        __syncthreads();
    }
    if (t == 0) {
        float s0 = red[0] + fc2b[0];
        float mx = s0;                              // max over size-1 axis
        out[g] = (s0 - mx) - logf(expf(s0 - mx));   // == 0 for finite inputs
    }
}

// ---------------------------------------------------------------------------
extern "C" void kernel_launch(void* const* d_in, const int* in_sizes, int n_in,
                              void* d_out, int out_size, void* d_ws, size_t ws_size,
                              hipStream_t stream) {
    (void)n_in; (void)ws_size;
    const float* x     = (const float*)d_in[0];
    const int*   ei    = (const int*)d_in[1];
    const float* ea    = (const float*)d_in[2];
    const int*   batch = (const int*)d_in[3];
    const float* W1a   = (const float*)d_in[4];
    const float* b1a   = (const float*)d_in[5];
    const float* W1b   = (const float*)d_in[6];
    const float* b1b   = (const float*)d_in[7];
    const float* root1 = (const float*)d_in[8];
    const float* bias1 = (const float*)d_in[9];
    const float* W2a   = (const float*)d_in[10];
    const float* b2a   = (const float*)d_in[11];
    const float* W2b   = (const float*)d_in[12];
    const float* b2b   = (const float*)d_in[13];
    const float* root2 = (const float*)d_in[14];
    const float* bias2 = (const float*)d_in[15];
    const float* fc1W  = (const float*)d_in[16];
    const float* fc1b  = (const float*)d_in[17];
    const float* fc2W  = (const float*)d_in[18];
    const float* fc2b  = (const float*)d_in[19];

    const int Nn = in_sizes[0] / 126;
    const int E  = in_sizes[2] / 19;
    const int G  = out_size;          // output is [G, 1]
    const int* src  = ei;
    const int* dstp = ei + E;

    constexpr int WT1 = 26 * 128 * 32;   // swizzled conv1 weights (padded)
    constexpr int WT2 = 26 * 32 * 64;    // swizzled conv2 weights

    // Workspace layout (floats). Atomic-accumulated buffers first (zeroed).
    float* ws   = (float*)d_ws;
    size_t off  = 0;
    float* agg1 = ws + off; off += (size_t)Nn * 32;
    float* agg2 = ws + off; off += (size_t)Nn * 64;
    float* cntv = ws + off; off += (size_t)Nn;
    float* gsum = ws + off; off += (size_t)G * 64;
    float* gcnt = ws + off; off += (size_t)G;
    off = (off + 3) & ~(size_t)3;          // 16B-align remaining buffers
    const size_t zeroN = off;
    float* T1  = ws + off; off += (size_t)E * 25;
    float* T2  = ws + off; off += (size_t)E * 25;
    float* h1  = ws + off; off += (size_t)Nn * 32;
    float* h2  = ws + off; off += (size_t)Nn * 64;
    off = (off + 3) & ~(size_t)3;
    float* Wt1 = ws + off; off += WT1;
    float* Wt2 = ws + off; off += WT2;

    zero_kernel<<<(int)((zeroN + 255) / 256), 256, 0, stream>>>(ws, (int)zeroN);

    // One-time weight swizzles into WMMA B-fragment order (L2-resident after).
    swizzle_wb_kernel<126, 32><<<(WT1 + 255) / 256, 256, 0, stream>>>(W1b, b1b, Wt1, WT1);
    swizzle_wb_kernel<32, 64><<<(WT2 + 255) / 256, 256, 0, stream>>>(W2b, b2b, Wt2, WT2);

    edge_mlp_count_kernel<<<(E + 255) / 256, 256, 0, stream>>>(
        ea, W1a, b1a, W2a, b2a, dstp, T1, T2, cntv, E);

    // conv1 messages: 25 slabs of K=128 (+bias slab), 2 output tiles -> 2 waves
    nnconv_msg_wmma_kernel<126, 32><<<(E + 15) / 16, 64, 0, stream>>>(
        x, T1, Wt1, src, dstp, agg1, E);

    node_update_kernel<126, 32><<<(Nn * 32 + 255) / 256, 256, 0, stream>>>(
        x, root1, bias1, agg1, cntv, h1, Nn);

    // conv2 messages: 25 slabs of K=32 (+bias slab), 4 output tiles -> 4 waves
    nnconv_msg_wmma_kernel<32, 64><<<(E + 15) / 16, 128, 0, stream>>>(
        h1, T2, Wt2, src, dstp, agg2, E);

    node_update_kernel<32, 64><<<(Nn * 64 + 255) / 256, 256, 0, stream>>>(
        h1, root2, bias2, agg2, cntv, h2, Nn);

    pool_kernel<<<(Nn * 64 + 255) / 256, 256, 0, stream>>>(h2, batch, gsum, gcnt, Nn);

    readout_kernel<<<G, 128, 0, stream>>>(gsum, gcnt, fc1W, fc1b, fc2W, fc2b,
                                          (float*)d_out);
}